// ObjectDetection_7344394076309
// MI455X (gfx1250) — compile-verified
//
#include <hip/hip_runtime.h>

typedef __attribute__((ext_vector_type(2))) float v2f;
typedef __attribute__((ext_vector_type(8))) float v8f;

#define BATCH 8
#define NANCH 65536            // anchors per image (= 128*128*4), power of two
#define NCLS  80
#define NBOX  32
#define POS_T 0.5f
#define NEG_T 0.1f
#define WAVES_PER_BLOCK 8      // 256 threads = 8 wave32
#define TILE 16                // anchors per wave tile (WMMA M)
#define ANCH_PER_BLOCK (WAVES_PER_BLOCK * TILE)          // 128
#define NBLOCKS ((BATCH * NANCH) / ANCH_PER_BLOCK)       // 4096
#define BLOCKS_PER_IMG (NANCH / ANCH_PER_BLOCK)          // 512
#define KSTEPS (NCLS / 4)                                // 20

#define LOG2E 1.4426950408889634f
#define LN2   0.6931471805599453f

__global__ __launch_bounds__(256) void det_loss_main(
    const float* __restrict__ logits,      // (B, A, C)
    const float* __restrict__ anchors,     // (A, 4)
    const float* __restrict__ boxes,       // (B, M, 4)
    const int*   __restrict__ labels_idx,  // (B, M)
    float* __restrict__ out_conf,          // (B*A)
    float* __restrict__ out_idx,           // (B*A) stored as float
    float* __restrict__ losspart,          // (NBLOCKS)
    float* __restrict__ pospart)           // (NBLOCKS)
{
  __shared__ float s_wloss[WAVES_PER_BLOCK];
  __shared__ int   s_pos;

  const int tid  = threadIdx.x;
  const int lane = tid & 31;
  const int wave = tid >> 5;
  const int row  = lane & 15;   // anchor row within tile
  const int half = lane >> 4;   // which K pair this lane feeds to WMMA

  if (tid == 0) s_pos = 0;
  __syncthreads();

  const int b    = blockIdx.x / BLOCKS_PER_IMG;           // scalar: s_load of boxes
  const int tile = blockIdx.x * WAVES_PER_BLOCK + wave;
  const int g    = tile * TILE + row;     // global anchor id (dup across lane halves)
  const int a    = g & (NANCH - 1);

  // ---- anchor<->gt-box assignment, split across lane halves:
  //      lower half scans boxes 0..15, upper half 16..31, combine via xor-16 ----
  const float4 an = ((const float4*)anchors)[a];
  const float areaA = (an.z - an.x) * (an.w - an.y);
  float best = 0.0f; int bi = 16 * half;
  #pragma unroll
  for (int j0 = 0; j0 < NBOX / 2; ++j0) {
    const int j = 16 * half + j0;
    const float4 bx = ((const float4*)boxes)[b * NBOX + j];
    float ltx = fmaxf(an.x, bx.x), lty = fmaxf(an.y, bx.y);
    float rbx = fminf(an.z, bx.z), rby = fminf(an.w, bx.w);
    float w  = fmaxf(rbx - ltx, 0.0f), h = fmaxf(rby - lty, 0.0f);
    float inter = w * h;
    float areaB = (bx.z - bx.x) * (bx.w - bx.y);
    float uni   = areaA + areaB - inter;
    float iou   = (uni > 0.0f) ? (inter * __builtin_amdgcn_rcpf(uni)) : 0.0f;
    if (iou > best) { best = iou; bi = j; }                 // first-max tie rule
  }
  {
    float ob = __shfl_xor(best, 16, 32);
    int   oj = __shfl_xor(bi,   16, 32);
    if (ob > best || (ob == best && oj < bi)) { best = ob; bi = oj; }
  }
  int code;                              // >=0: pos target class; -1: neg; -2: ignore
  if (best >= POS_T)      code = labels_idx[b * NBOX + bi];
  else if (best < NEG_T)  code = -1;
  else                    code = -2;
  const float lab = (code >= -1) ? 1.0f : 0.0f;   // "labeled" row mask
  const int   pos = (best >= POS_T) ? 1 : 0;

  // ---- stage the whole 16x80 tile's lane-share up front (20 x b64 loads) ----
  const float* row_ptr = logits + (size_t)g * NCLS + 2 * half;
  v2f xs[KSTEPS];
  #pragma unroll
  for (int i = 0; i < KSTEPS; ++i)
    xs[i] = *(const v2f*)(row_ptr + 4 * i);

  // ---- focal-BCE; row reduction via V_WMMA_F32_16X16X4_F32 ----
  v8f c = {};                               // D: every column = masked row loss
  const v2f ones = { 1.0f, 1.0f };          // all-ones B => D[m][n] = rowsum(m)
  float bm = -3.4e38f; int bmi = 0;         // max-logit (== argmax of sigmoid)

  #pragma unroll
  for (int i = 0; i < KSTEPS; ++i) {
    const int cls = 4 * i + 2 * half;       // ISA A-fragment: VGPR0=K{0|2}, VGPR1=K{1|3}
    const v2f x2 = xs[i];

    if (x2.x > bm) { bm = x2.x; bmi = cls; }
    if (x2.y > bm) { bm = x2.y; bmi = cls + 1; }

    // raw HW trans: 3 per element (exp2, rcp, log2), no libm guard code.
    //   e = 2^(-|x|*log2e);  q = 1/(1+e);  sigmoid = x>=0 ? q : 1-q
    //   softplus(-|x|) = log2(1+e) * ln2
    float e0 = __builtin_amdgcn_exp2f(-LOG2E * fabsf(x2.x));
    float e1 = __builtin_amdgcn_exp2f(-LOG2E * fabsf(x2.y));
    float d0 = 1.0f + e0;
    float d1 = 1.0f + e1;
    float q0 = __builtin_amdgcn_rcpf(d0);
    float q1 = __builtin_amdgcn_rcpf(d1);
    float p0 = (x2.x >= 0.0f) ? q0 : (1.0f - q0);
    float p1 = (x2.y >= 0.0f) ? q1 : (1.0f - q1);
    float sp0 = LN2 * __builtin_amdgcn_logf(d0);
    float sp1 = LN2 * __builtin_amdgcn_logf(d1);
    float t0 = (cls     == code) ? 1.0f : 0.0f;
    float t1 = (cls + 1 == code) ? 1.0f : 0.0f;
    float bce0 = fmaxf(x2.x, 0.0f) - x2.x * t0 + sp0;
    float bce1 = fmaxf(x2.y, 0.0f) - x2.y * t1 + sp1;
    float w0 = t0 - p0; w0 *= w0;
    float w1 = t1 - p1; w1 *= w1;

    v2f av;
    av.x = lab * bce0 * w0;
    av.y = lab * bce1 * w1;
    // 8 args: (neg_a, A, neg_b, B, c_mod, C, reuse_a, reuse_b)
    c = __builtin_amdgcn_wmma_f32_16x16x4_f32(false, av, false, ones,
                                              (short)0, c, false, false);
  }

  // Tile loss = sum over 16 rows. Lane n(<16): c[0..7] = rows 0..7 (col n);
  // lane n+16: rows 8..15. One xor-16 combine -> full tile loss on every lane.
  float s = c[0]+c[1]+c[2]+c[3]+c[4]+c[5]+c[6]+c[7];
  s += __shfl_xor(s, 16, 32);

  // positive-anchor count (each row counted once: lower half of the ballot)
  unsigned long long pb = __ballot(pos != 0);
  int pcnt = __popcll(pb & 0xFFFFull);

  if (lane == 0) {
    s_wloss[wave] = s;
    atomicAdd(&s_pos, pcnt);                // integer LDS atomic: deterministic
  }

  // per-anchor conf / argmax outputs (combine the two class-subsets)
  float om = __shfl_xor(bm, 16, 32);
  int   oi = __shfl_xor(bmi, 16, 32);
  if (om > bm || (om == bm && oi < bmi)) { bm = om; bmi = oi; }
  if (lane < 16) {
    float ec = __builtin_amdgcn_exp2f(-LOG2E * fabsf(bm));
    float rc = __builtin_amdgcn_rcpf(1.0f + ec);
    out_conf[g] = (bm >= 0.0f) ? rc : (1.0f - rc);
    out_idx[g]  = (float)bmi;
  }

  __syncthreads();
  if (tid == 0) {
    float ls = 0.0f;
    #pragma unroll
    for (int wv = 0; wv < WAVES_PER_BLOCK; ++wv) ls += s_wloss[wv];
    losspart[blockIdx.x] = ls;
    pospart[blockIdx.x]  = (float)s_pos;
  }
}

__global__ __launch_bounds__(256) void det_loss_finalize(
    const float* __restrict__ losspart,
    const float* __restrict__ pospart,
    float* __restrict__ out0)
{
  __shared__ float sl[256], sp[256];
  const int tid = threadIdx.x;
  float l = 0.0f, p = 0.0f;
  for (int i = tid; i < NBLOCKS; i += 256) { l += losspart[i]; p += pospart[i]; }
  sl[tid] = l; sp[tid] = p;
  __syncthreads();
  for (int st = 128; st > 0; st >>= 1) {   // fixed-order tree: deterministic
    if (tid < st) { sl[tid] += sl[tid + st]; sp[tid] += sp[tid + st]; }
    __syncthreads();
  }
  if (tid == 0) out0[0] = sl[0] / sp[0];
}

extern "C" void kernel_launch(void* const* d_in, const int* in_sizes, int n_in,
                              void* d_out, int out_size, void* d_ws, size_t ws_size,
                              hipStream_t stream) {
  const float* logits     = (const float*)d_in[0];
  const float* anchors    = (const float*)d_in[1];
  const float* boxes      = (const float*)d_in[2];
  /* d_in[3] = labels one-hot: redundant with labels_idx, unused */
  const int*   labels_idx = (const int*)d_in[4];

  float* out      = (float*)d_out;
  float* out_loss = out;                          // [0]
  float* out_conf = out + 1;                      // [1 .. B*A]
  float* out_idx  = out + 1 + BATCH * NANCH;      // [1+B*A .. 1+2*B*A]

  float* losspart = (float*)d_ws;                 // NBLOCKS floats
  float* pospart  = losspart + NBLOCKS;           // NBLOCKS floats (32 KB total)

  det_loss_main<<<dim3(NBLOCKS), dim3(256), 0, stream>>>(
      logits, anchors, boxes, labels_idx, out_conf, out_idx, losspart, pospart);
  det_loss_finalize<<<dim3(1), dim3(256), 0, stream>>>(losspart, pospart, out_loss);
}